// StillingerWeberLayer_82291573391482
// MI455X (gfx1250) — compile-verified
//
#include <hip/hip_runtime.h>

// Stillinger-Weber forces on MI455X (gfx1250, wave32).
// One wave32 per atom (20000 waves -> fills the WGP array). Lanes 0..15
// compute the 16 two-body terms; a single V_WMMA_F32_16X16X4_F32 computes
// the 16x16 Gram matrix of the neighbor displacement vectors (R * R^T,
// K=3 padded to 4), which supplies r_jk^2 for all 120 three-body pairs;
// all 32 lanes then sweep the 120 (j,k) pairs in 4 fully-unrolled sweeps
// using ds_bpermute (__shfl) gathers. All data is L2-resident (<3 MB);
// scatter-adds are global f32 atomics served by the 192MB L2.

#define N_ATOMS 20000
#define MAX_NBR 16
#define NL_W (MAX_NBR + 1)
#define WAVES_PER_BLOCK 8
#define BLOCK (WAVES_PER_BLOCK * 32)
#define N_TRIPLE 120

typedef float v2f __attribute__((ext_vector_type(2)));
typedef float v8f __attribute__((ext_vector_type(8)));

// (j,k) upper-triangle pair table, packed (j<<8)|k, built at compile time.
struct TriTab { unsigned short v[128]; };
static constexpr TriTab make_tri() {
    TriTab t{};
    int p = 0;
    for (int j = 0; j < MAX_NBR; ++j)
        for (int k = j + 1; k < MAX_NBR; ++k)
            t.v[p++] = (unsigned short)((j << 8) | k);
    for (; p < 128; ++p) t.v[p] = 0;
    return t;
}
__constant__ TriTab TRI = make_tri();

__device__ __forceinline__ float sel3(float a0, float a1, float a2, int i) {
    float r = (i == 1) ? a1 : a0;
    return (i == 2) ? a2 : r;
}
__device__ __forceinline__ float sel2(float a0, float a1, int i) {
    return i ? a1 : a0;
}

__global__ void sw_zero(float* __restrict__ out, int n) {
    int t = blockIdx.x * blockDim.x + threadIdx.x;
    if (t < n) out[t] = 0.0f;
}

__global__ void __launch_bounds__(BLOCK)
sw_kernel(const int* __restrict__ elements, const float* __restrict__ coords,
          const int* __restrict__ nl,
          const float* __restrict__ Ap, const float* __restrict__ Bp,
          const float* __restrict__ Pp, const float* __restrict__ Qp,
          const float* __restrict__ SIp, const float* __restrict__ GAp,
          const float* __restrict__ CUp, const float* __restrict__ LAp,
          const float* __restrict__ CBp, const float* __restrict__ CJp,
          float* __restrict__ out) {
    __shared__ float sG[WAVES_PER_BLOCK][256];  // 16x16 Gram matrix per wave

    const int lane = threadIdx.x & 31;
    const int wv   = threadIdx.x >> 5;
    const int i    = blockIdx.x * WAVES_PER_BLOCK + wv;   // atom index, exact cover

    float* F = out + 1;            // forces; out[0] = energy
    // ---- parameters (uniform scalar loads) ----
    const float A0 = Ap[0],  A1 = Ap[1],  A2 = Ap[2];
    const float B0 = Bp[0],  B1 = Bp[1],  B2 = Bp[2];
    const float P0 = Pp[0],  P1 = Pp[1],  P2 = Pp[2];
    const float Q0 = Qp[0],  Q1 = Qp[1],  Q2 = Qp[2];
    const float S0 = SIp[0], S1 = SIp[1], S2 = SIp[2];
    const float G0 = GAp[0], G1 = GAp[1], G2 = GAp[2];
    const float C0 = CUp[0], C1 = CUp[1], C2 = CUp[2];
    const float L0 = LAp[0], L1 = LAp[1];
    const float CB0 = CBp[0], CB1 = CBp[1];
    const float CJ0 = CJp[0], CJ1 = CJp[1];

    const int e_i = elements[i * NL_W];
    const float xi = coords[3 * i + 0];
    const float yi = coords[3 * i + 1];
    const float zi = coords[3 * i + 2];

    // center-atom force accumulators (reduced across wave at end)
    float cfx = 0.0f, cfy = 0.0f, cfz = 0.0f;

    // per-neighbor state held by lanes 0..15
    float rvx = 0.0f, rvy = 0.0f, rvz = 0.0f, rr = 1.0f;
    int enb = 0;                       // (element << 16) | neighbor_index

    // ================= two-body phase (lanes 0..15, one neighbor each) ====
    if (lane < MAX_NBR) {
        const int j = lane;
        const int nbrIdx = nl[i * NL_W + 1 + j];
        const int en     = elements[i * NL_W + 1 + j];
        enb = (en << 16) | nbrIdx;
        const float xn = coords[3 * nbrIdx + 0];
        const float yn = coords[3 * nbrIdx + 1];
        const float zn = coords[3 * nbrIdx + 2];
        rvx = xn - xi; rvy = yn - yi; rvz = zn - zi;
        const float r2 = rvx * rvx + rvy * rvy + rvz * rvz;
        rr = sqrtf(fmaxf(r2, 1e-12f));

        const int ij = e_i + en;                 // 0..2
        const float c2  = sel3(C0, C1, C2, ij);
        const bool  m2  = (nbrIdx != i) && (rr < c2);
        const float rs  = m2 ? rr : 1.0f;
        const float drs = m2 ? (rs - c2) : -1.0f;
        const float sg  = sel3(S0, S1, S2, ij);
        const float sgr = sg / rs;
        const float pij = sel3(P0, P1, P2, ij);
        const float qij = sel3(Q0, Q1, Q2, ij);
        const float Bij = sel3(B0, B1, B2, ij);
        const float Aij = sel3(A0, A1, A2, ij);
        const float Bpq = Bij * powf(sgr, pij) - powf(sgr, qij);
        const float ex  = expf(sg / drs);
        const float E2  = m2 ? (Aij * Bpq * ex) : 0.0f;
        const float sd  = sg / drs;
        float F2 = qij * powf(sgr, qij + 1.0f) - pij * Bij * powf(sgr, pij + 1.0f)
                 - Bpq * sd * sd;
        F2 = m2 ? (F2 * Aij * ex / sg) : 0.0f;
        const float fs = 0.5f * F2 / rs;
        const float Fpx = fs * rvx, Fpy = fs * rvy, Fpz = fs * rvz;
        cfx += Fpx; cfy += Fpy; cfz += Fpz;
        if (F2 != 0.0f) {
            atomicAdd(&F[3 * nbrIdx + 0], -Fpx);
            atomicAdd(&F[3 * nbrIdx + 1], -Fpy);
            atomicAdd(&F[3 * nbrIdx + 2], -Fpz);
        }
        // energy = 0.5 * E2[-1,-1]: last atom, last neighbor only
        if (i == N_ATOMS - 1 && j == MAX_NBR - 1) out[0] = 0.5f * E2;
    }

    // ================= Gram matrix via WMMA (all 32 lanes, EXEC all 1s) ===
    // A (16x4 f32): lane L holds M=L&15; lanes<16 -> K={0,1}=(x,y),
    // lanes>=16 -> K={2,3}=(z,0).  B = A^T has the mirrored layout, so the
    // same register pair serves as the B fragment.  D = R*R^T (16x16 f32).
    {
        const int  m  = lane & 15;
        const bool hi = lane >= 16;
        const float xm = __shfl(rvx, m);
        const float ym = __shfl(rvy, m);
        const float zm = __shfl(rvz, m);
        v2f a; a.x = hi ? zm : xm; a.y = hi ? 0.0f : ym;
        v8f cacc = {0.f, 0.f, 0.f, 0.f, 0.f, 0.f, 0.f, 0.f};
        v8f g = __builtin_amdgcn_wmma_f32_16x16x4_f32(
            /*neg_a=*/false, a, /*neg_b=*/false, a,
            /*c_mod=*/(short)0, cacc, /*reuse_a=*/false, /*reuse_b=*/false);
        // D layout: VGPR v -> row (hi? v+8 : v), col = lane&15
        #pragma unroll
        for (int v = 0; v < 8; ++v) {
            const int row = hi ? (v + 8) : v;
            sG[wv][row * 16 + m] = g[v];
        }
    }
    __syncthreads();

    // ====== three-body phase: 120 pairs, 4 sweeps, fully unrolled =========
    // Unrolling breaks the load->bpermute->exp dependency chain between
    // sweeps: all 4 TRI loads and all gathers issue up front.
    #pragma unroll
    for (int it = 0; it < 4; ++it) {
        const int p = it * 32 + lane;
        const unsigned short jk = TRI.v[(p < N_TRIPLE) ? p : 0];
        const int j = jk >> 8, k = jk & 0xFF;

        const float rj = __shfl(rr, j),  rk = __shfl(rr, k);
        const float xj = __shfl(rvx, j), yj = __shfl(rvy, j), zj = __shfl(rvz, j);
        const float xk = __shfl(rvx, k), yk = __shfl(rvy, k), zk = __shfl(rvz, k);
        const int   pj = __shfl(enb, j), pk = __shfl(enb, k);
        const int   ej = pj >> 16,       ekk = pk >> 16;
        const int   idxj = pj & 0xFFFF,  idxk = pk & 0xFFFF;

        // r_jk^2 from the WMMA Gram matrix: |v_k - v_j|^2 = rj^2+rk^2-2*G[j][k]
        const float Gjk = sG[wv][j * 16 + k];
        const float rjk = sqrtf(fmaxf(rj * rj + rk * rk - 2.0f * Gjk, 1e-12f));
        const float rjkx = xk - xj, rjky = yk - yj, rjkz = zk - zj;

        const int ij_j = e_i + ej, ij_k = e_i + ekk;
        int tri = 2 - (e_i + ej + ekk);
        tri = tri < 0 ? 0 : (tri > 1 ? 1 : tri);
        const float cij = sel3(C0, C1, C2, ij_j);
        const float cik = sel3(C0, C1, C2, ij_k);
        const float gij = sel3(G0, G1, G2, ij_j);
        const float gik = sel3(G0, G1, G2, ij_k);

        const bool m3 = (p < N_TRIPLE) && (e_i != ej) && (ej == ekk) &&
                        (rj < cij) && (rk < cik) && (rjk < sel2(CJ0, CJ1, tri)) &&
                        (idxj != i) && (idxk != i);

        const float rijs = m3 ? rj : 1.0f;
        const float riks = m3 ? rk : 1.0f;
        const float rjks = m3 ? rjk : 1.0f;
        const float dijs = m3 ? (rijs - cij) : -1.0f;
        const float diks = m3 ? (riks - cik) : -1.0f;

        const float rij2 = rijs * rijs, rik2 = riks * riks, rjk2 = rjks * rjks;
        const float cosb = (rij2 + rik2 - rjk2) / (2.0f * rijs * riks);
        const float cd   = cosb - sel2(CB0, CB1, tri);
        const float ex3  = expf(gij / dijs + gik / diks);
        const float dij  = -gij / (dijs * dijs);
        const float dik  = -gik / (diks * diks);
        const float dc_rij = (rij2 - rik2 + rjk2) / (2.0f * rij2 * riks);
        const float dc_rik = (rik2 - rij2 + rjk2) / (2.0f * rik2 * rijs);
        const float dc_rjk = -rjks / (rijs * riks);

        const float base = m3 ? (sel2(L0, L1, tri) * cd * ex3) : 0.0f;
        const float d0 = base * (dij * cd + 2.0f * dc_rij);
        const float d1 = base * (dik * cd + 2.0f * dc_rik);
        const float d2 = base * (2.0f * dc_rjk);

        const float s0 = d0 / rijs, s1 = d1 / riks, s2 = d2 / rjks;
        const float Fijx = s0 * xj,   Fijy = s0 * yj,   Fijz = s0 * zj;
        const float Fikx = s1 * xk,   Fiky = s1 * yk,   Fikz = s1 * zk;
        const float Fjkx = s2 * rjkx, Fjky = s2 * rjky, Fjkz = s2 * rjkz;

        cfx += Fijx + Fikx; cfy += Fijy + Fiky; cfz += Fijz + Fikz;
        if (m3) {
            atomicAdd(&F[3 * idxj + 0], Fjkx - Fijx);
            atomicAdd(&F[3 * idxj + 1], Fjky - Fijy);
            atomicAdd(&F[3 * idxj + 2], Fjkz - Fijz);
            atomicAdd(&F[3 * idxk + 0], -(Fikx + Fjkx));
            atomicAdd(&F[3 * idxk + 1], -(Fiky + Fjky));
            atomicAdd(&F[3 * idxk + 2], -(Fikz + Fjkz));
        }
    }

    // ================= center-atom force: wave32 reduction ================
    #pragma unroll
    for (int off = 16; off > 0; off >>= 1) {
        cfx += __shfl_xor(cfx, off);
        cfy += __shfl_xor(cfy, off);
        cfz += __shfl_xor(cfz, off);
    }
    if (lane == 0) {
        atomicAdd(&F[3 * i + 0], cfx);
        atomicAdd(&F[3 * i + 1], cfy);
        atomicAdd(&F[3 * i + 2], cfz);
    }
}

extern "C" void kernel_launch(void* const* d_in, const int* in_sizes, int n_in,
                              void* d_out, int out_size, void* d_ws, size_t ws_size,
                              hipStream_t stream) {
    const int*   elements = (const int*)d_in[0];
    const float* coords   = (const float*)d_in[1];
    const int*   nl       = (const int*)d_in[2];
    // d_in[3] = padding (size 0)
    const float* A   = (const float*)d_in[4];
    const float* B   = (const float*)d_in[5];
    const float* P   = (const float*)d_in[6];
    const float* Q   = (const float*)d_in[7];
    const float* SIG = (const float*)d_in[8];
    const float* GAM = (const float*)d_in[9];
    const float* CUT = (const float*)d_in[10];
    const float* LAM = (const float*)d_in[11];
    const float* CB  = (const float*)d_in[12];
    const float* CJK = (const float*)d_in[13];
    float* out = (float*)d_out;   // [0]=energy, [1..60000]=F

    sw_zero<<<(out_size + 255) / 256, 256, 0, stream>>>(out, out_size);
    sw_kernel<<<N_ATOMS / WAVES_PER_BLOCK, BLOCK, 0, stream>>>(
        elements, coords, nl, A, B, P, Q, SIG, GAM, CUT, LAM, CB, CJK, out);
}